// OnlineHadamard_66279935312240
// MI455X (gfx1250) — compile-verified
//
#include <hip/hip_runtime.h>

typedef __attribute__((ext_vector_type(2))) float v2f;
typedef __attribute__((ext_vector_type(4))) float v4f;
typedef __attribute__((ext_vector_type(8))) float v8f;

#define DIM 4096
#define ROWS_PER_BLOCK 8
// Pad every 16-float group by 4 floats: keeps b128 alignment on fills and
// makes the stride-16 pass LDS-bank-conflict-free (20n mod 64 distinct).
#define PADF(a) ((a) + ((((a) >> 4)) << 2))
#define LDS_ROW_F (DIM + (DIM / 16) * 4)   // 5120 floats = 20 KiB per row

// One pass of H16 applied along the axis with element-stride S, batching 16
// elements of the axis with stride T, tile bases at t*BSTRIDE.
// Computes (per tile) D[m][n] = sum_k H[m][k] * X[k][n], in place in LDS.
template <int S, int T, int BSTRIDE>
__device__ __forceinline__ void had_pass(float* __restrict__ row, int lane,
                                         const v2f (&afrag)[4]) {
  const int n = lane & 15;
  const int h = lane >> 4;
  const int lo = n * T + 2 * h * S;  // per-lane invariant part of B address
  #pragma unroll
  for (int t = 0; t < 16; ++t) {
    const int base = t * BSTRIDE;
    v8f acc = {};
    #pragma unroll
    for (int c = 0; c < 4; ++c) {
      // B fragment (4x16 f32, 2 VGPRs): VGPR0 = rows {4c+0 | 4c+2} across
      // lane halves, VGPR1 = rows {4c+1 | 4c+3}; N = lane & 15.
      const int a0 = base + lo + 4 * c * S;
      v2f b;
      b.x = row[PADF(a0)];
      b.y = row[PADF(a0 + S)];
      acc = __builtin_amdgcn_wmma_f32_16x16x4_f32(
          /*neg_a=*/false, afrag[c], /*neg_b=*/false, b,
          /*c_mod=*/(short)0, acc, /*reuse_a=*/false, /*reuse_b=*/false);
    }
    // D fragment (16x16 f32, 8 VGPRs): VGPR v -> M = v + 8*h, N = lane & 15.
    #pragma unroll
    for (int v = 0; v < 8; ++v) {
      const int m = v + 8 * h;
      row[PADF(base + m * S + n * T)] = acc[v];
    }
  }
}

// Branchless Hadamard sign: +1.0f / -1.0f, no vcc usage.
__device__ __forceinline__ float hsign(int m, int k) {
  const unsigned parity = (unsigned)__popc(m & k);        // low bit = parity
  return __uint_as_float((parity << 31) | 0x3f800000u);   // sign | 1.0f
}

__global__ __launch_bounds__(32 * ROWS_PER_BLOCK)
void fwht4096_kernel(const float* __restrict__ x, float* __restrict__ y,
                     int nrows) {
  extern __shared__ float lds[];
  const int wave = threadIdx.x >> 5;
  const int lane = threadIdx.x & 31;
  float* row = lds + wave * LDS_ROW_F;

  const int gr = blockIdx.x * ROWS_PER_BLOCK + wave;  // wave-uniform
  if (gr >= nrows) return;                            // whole-wave exit; EXEC stays all-1s

  const float* __restrict__ src = x + (size_t)gr * DIM;
  float* __restrict__ dst = y + (size_t)gr * DIM;

  // -------- Global -> LDS (coalesced float4, padded store stays 16B-aligned)
  #pragma unroll
  for (int i = 0; i < DIM / 128; ++i) {
    const int a = i * 128 + lane * 4;   // multiple of 4, within a 16-group
    v4f v = *(const v4f*)(src + a);
    *(v4f*)(row + PADF(a)) = v;
  }

  // -------- A fragments = H16 column chunks (H is symmetric, built once)
  // A (16x4 f32, 2 VGPRs): M = lane & 15; lanes<16 hold K = 4c+{0,1},
  // lanes>=16 hold K = 4c+{2,3}.  H[m][k] = parity(m & k) ? -1 : +1.
  v2f afrag[4];
  {
    const int m = lane & 15;
    const int h = lane >> 4;
    #pragma unroll
    for (int c = 0; c < 4; ++c) {
      const int k0 = 4 * c + 2 * h;
      afrag[c].x = hsign(m, k0);
      afrag[c].y = hsign(m, k0 + 1);
    }
  }
  // Pin the fragments in VGPRs: the opaque asm prevents the compiler from
  // rematerializing the popcount/sign computation inside the unrolled
  // WMMA loops (observed in round-1 asm as repeated v_bcnt/v_cndmask).
  #pragma unroll
  for (int c = 0; c < 4; ++c) {
    asm volatile("" : "+v"(afrag[c]));
  }

  // -------- Three Kronecker passes: index e = d2*256 + d1*16 + d0
  // Row is wave-private; per-wave DS ordering guarantees pass-to-pass visibility.
  had_pass<1,   16, 256>(row, lane, afrag);  // H along d0, batch over d1, slice d2
  had_pass<16,   1, 256>(row, lane, afrag);  // H along d1, batch over d0, slice d2
  had_pass<256,  1,  16>(row, lane, afrag);  // H along d2, batch over d0, slice d1

  // -------- LDS -> Global with 1/sqrt(4096) = 1/64 scale, coalesced float4
  const float scale = 0.015625f;
  #pragma unroll
  for (int i = 0; i < DIM / 128; ++i) {
    const int a = i * 128 + lane * 4;
    v4f v = *(const v4f*)(row + PADF(a));
    v.x *= scale; v.y *= scale; v.z *= scale; v.w *= scale;
    *(v4f*)(dst + a) = v;
  }
}

extern "C" void kernel_launch(void* const* d_in, const int* in_sizes, int n_in,
                              void* d_out, int out_size, void* d_ws, size_t ws_size,
                              hipStream_t stream) {
  const float* x = (const float*)d_in[0];
  float* y = (float*)d_out;
  const int rows = in_sizes[0] / DIM;  // 4*2048 = 8192
  const dim3 block(32 * ROWS_PER_BLOCK);
  const dim3 grid((rows + ROWS_PER_BLOCK - 1) / ROWS_PER_BLOCK);
  const size_t shmem = (size_t)ROWS_PER_BLOCK * LDS_ROW_F * sizeof(float); // 160 KiB
  hipLaunchKernelGGL(fwht4096_kernel, grid, block, shmem, stream, x, y, rows);
}